// OneConv_21337397526624
// MI455X (gfx1250) — compile-verified
//
#include <hip/hip_runtime.h>
#include <hip/hip_bf16.h>
#include <math.h>

typedef float v2f __attribute__((ext_vector_type(2)));
typedef float v8f __attribute__((ext_vector_type(8)));

#define FIN  48
#define COUT 16
#define HID  8

// ---------------------------------------------------------------------------
// Kernel 1: y = x @ W  via V_WMMA_F32_16X16X4_F32 (fp32 exact, matches ref)
// One wave per 16-row tile; 12 WMMAs over K=48.
// A 16x4 layout: lane m=l%16, half=l/16; V0=K(2*half), V1=K(2*half+1)
// B 4x16 layout mirrored; D: VGPR j holds M=j+8*half, N=lane%16.
// ---------------------------------------------------------------------------
__global__ void __launch_bounds__(256)
feast_gemm_xw(const float* __restrict__ x, const float* __restrict__ W,
              float* __restrict__ y, int N, int ntiles) {
  int lane = threadIdx.x & 31;
  int wave = threadIdx.x >> 5;
  int tile = blockIdx.x * 8 + wave;
  if (tile >= ntiles) return;          // wave-uniform: EXEC stays all-1s
  int m    = lane & 15;
  int half = lane >> 4;
  int row  = tile * 16 + m;
  if (row >= N) row = N - 1;           // clamp loads, keep EXEC full
  v8f acc = {0.f, 0.f, 0.f, 0.f, 0.f, 0.f, 0.f, 0.f};
#pragma unroll
  for (int k = 0; k < FIN; k += 4) {
    int ka = k + 2 * half;
    v2f a, b;
    a[0] = x[row * FIN + ka + 0];
    a[1] = x[row * FIN + ka + 1];
    b[0] = W[(ka + 0) * COUT + m];
    b[1] = W[(ka + 1) * COUT + m];
    acc = __builtin_amdgcn_wmma_f32_16x16x4_f32(
        /*neg_a=*/false, a, /*neg_b=*/false, b,
        /*c_mod=*/(short)0, acc, /*reuse_a=*/false, /*reuse_b=*/false);
  }
  int base = tile * 16 + 8 * half;
  if (tile * 16 + 16 <= N) {
    // Full tile (always the case for N % 16 == 0): unconditional stores.
#pragma unroll
    for (int j = 0; j < 8; ++j) y[(base + j) * COUT + m] = acc[j];
  } else {
    // Ragged final tile: per-row guard.
#pragma unroll
    for (int j = 0; j < 8; ++j)
      if (base + j < N) y[(base + j) * COUT + m] = acc[j];
  }
}

// ---------------------------------------------------------------------------
// Kernel 2: init — self-loop contribution: agg = y, deg = 1, loss accum = 0
// ---------------------------------------------------------------------------
__global__ void feast_init(const float* __restrict__ y, float* __restrict__ agg,
                           float* __restrict__ deg, float* __restrict__ lossAcc,
                           int N) {
  int i = blockIdx.x * 256 + threadIdx.x;
  if (i < N * COUT) agg[i] = y[i];
  if (i < N)        deg[i] = 1.0f;
  if (i == 0)       *lossAcc = 0.0f;
}

// ---------------------------------------------------------------------------
// Kernel 3: edge scatter — 16 threads per edge; coalesced 64B gather of
// y[src], 64B of channel-contiguous f32 atomics into agg[dst] (L2-resident).
// ---------------------------------------------------------------------------
__global__ void __launch_bounds__(256)
feast_scatter(const int* __restrict__ ei, const float* __restrict__ y,
              float* __restrict__ agg, float* __restrict__ deg, int E) {
  int t = blockIdx.x * 256 + threadIdx.x;
  int e = t >> 4;
  int c = t & 15;
  if (e >= E) return;
  int s = ei[e];          // edge_index[0][e]
  int d = ei[E + e];      // edge_index[1][e]
  atomicAdd(&agg[d * COUT + c], y[s * COUT + c]);
  if (c == 0) atomicAdd(&deg[d], 1.0f);
}

// ---------------------------------------------------------------------------
// Kernel 4: per-node head — mean+bias, ReLU, 16->8 MLP, ReLU, 8->1, sigmoid,
// weighted BCE; block tree-reduce, one atomicAdd per block into lossAcc.
// ---------------------------------------------------------------------------
__global__ void __launch_bounds__(256)
feast_head(const float* __restrict__ agg, const float* __restrict__ deg,
           const float* __restrict__ bias, const float* __restrict__ lin1_w,
           const float* __restrict__ lin1_b, const float* __restrict__ out_w,
           const float* __restrict__ out_b, const float* __restrict__ labels,
           const float* __restrict__ weights, float* __restrict__ d_out,
           float* __restrict__ lossAcc, int N) {
  __shared__ float red[256];
  int n = blockIdx.x * 256 + threadIdx.x;
  float contrib = 0.0f;
  if (n < N) {
    float invd = 1.0f / deg[n];   // deg >= 1 (self-loop) => max(deg,1)==deg
    float a[COUT];
#pragma unroll
    for (int c2 = 0; c2 < COUT; ++c2) {
      float v = agg[n * COUT + c2] * invd + bias[c2];
      a[c2] = v > 0.f ? v : 0.f;
    }
    float z = out_b[0];
#pragma unroll
    for (int j = 0; j < HID; ++j) {
      float h = lin1_b[j];
#pragma unroll
      for (int c2 = 0; c2 < COUT; ++c2) h += a[c2] * lin1_w[c2 * HID + j];
      h = h > 0.f ? h : 0.f;
      z += h * out_w[j];
    }
    float p = 1.0f / (1.0f + expf(-z));
    d_out[1 + n] = p;
    const float eps = 1e-7f;
    float pc  = fminf(fmaxf(p, eps), 1.0f - eps);
    float lab = labels[n];
    float bce = -(lab * logf(pc) + (1.0f - lab) * logf(1.0f - pc));
    contrib = weights[n] * bce;
  }
  red[threadIdx.x] = contrib;
  __syncthreads();
  for (int s = 128; s > 0; s >>= 1) {
    if ((int)threadIdx.x < s) red[threadIdx.x] += red[threadIdx.x + s];
    __syncthreads();
  }
  if (threadIdx.x == 0) atomicAdd(lossAcc, red[0]);
}

// ---------------------------------------------------------------------------
// Kernel 5: loss = acc / N
// ---------------------------------------------------------------------------
__global__ void feast_finalize(const float* __restrict__ lossAcc,
                               float* __restrict__ d_out, int N) {
  if (threadIdx.x == 0 && blockIdx.x == 0) d_out[0] = *lossAcc / (float)N;
}

// ---------------------------------------------------------------------------
extern "C" void kernel_launch(void* const* d_in, const int* in_sizes, int n_in,
                              void* d_out, int out_size, void* d_ws,
                              size_t ws_size, hipStream_t stream) {
  const float* x       = (const float*)d_in[0];
  const int*   ei      = (const int*)d_in[1];   // edge_index [2, E]
  const float* labels  = (const float*)d_in[2];
  const float* weights = (const float*)d_in[3];
  const float* W       = (const float*)d_in[4];
  // d_in[5] = u, d_in[6] = c : unused — softmax over H=1 is identically 1.0
  const float* bias    = (const float*)d_in[7];
  const float* lin1_w  = (const float*)d_in[8];
  const float* lin1_b  = (const float*)d_in[9];
  const float* out_w   = (const float*)d_in[10];
  const float* out_b   = (const float*)d_in[11];
  float* out = (float*)d_out;   // [0] = loss, [1..N] = p

  int N = in_sizes[0] / FIN;
  int E = in_sizes[1] / 2;

  // Workspace layout (all fp32): y[N*16] | agg[N*16] | deg[N] | lossAcc[1]
  float* y       = (float*)d_ws;
  float* agg     = y + (size_t)N * COUT;
  float* deg     = agg + (size_t)N * COUT;
  float* lossAcc = deg + N;

  int ntiles = (N + 15) / 16;
  feast_gemm_xw<<<(ntiles + 7) / 8, 256, 0, stream>>>(x, W, y, N, ntiles);

  int initThreads = N * COUT;
  feast_init<<<(initThreads + 255) / 256, 256, 0, stream>>>(y, agg, deg,
                                                            lossAcc, N);

  long long scatterThreads = (long long)E * COUT;
  feast_scatter<<<(unsigned)((scatterThreads + 255) / 256), 256, 0, stream>>>(
      ei, y, agg, deg, E);

  feast_head<<<(N + 255) / 256, 256, 0, stream>>>(agg, deg, bias, lin1_w,
                                                  lin1_b, out_w, out_b, labels,
                                                  weights, out, lossAcc, N);

  feast_finalize<<<1, 1, 0, stream>>>(lossAcc, out, N);
}